// GATExplainer_32985348833758
// MI455X (gfx1250) — compile-verified
//
#include <hip/hip_runtime.h>
#include <math.h>

// ---------------------------------------------------------------------------
// GATExplainer for MI455X (gfx1250, wave32, WMMA).
// Dense GEMMs use V_WMMA_F32_16X16X4_F32 (fp32 path, reference precision),
// with double-buffered async global->LDS staging (ASYNCcnt) when available.
// Scatter phases use f32 global atomics (dest arrays fit in 192MB L2).
// ---------------------------------------------------------------------------

typedef float v2f __attribute__((ext_vector_type(2)));
typedef float v8f __attribute__((ext_vector_type(8)));
typedef int   v4i __attribute__((ext_vector_type(4)));

// Problem constants (fixed by the reference's setup_inputs()).
#define NN      50000
#define EE      400000
#define FIN     262
#define HID     128
#define NHEAD   4
#define F1      (NHEAD * HID)   // 512
#define NGRAPH  16

// ---- CDNA5 async global->LDS copy (probe via __has_builtin) ---------------
#if defined(__has_builtin)
#if __has_builtin(__builtin_amdgcn_global_load_async_to_lds_b128)
#define ASYNC_LDS 1
#endif
#endif
#ifndef ASYNC_LDS
#define ASYNC_LDS 0
#endif

#if ASYNC_LDS
// Builtin signature (from hipcc diagnostic): (v4i AS1*, v4i AS3*, imm, imm).
__device__ __forceinline__ void async_cp16(const void* g, void* l) {
  __builtin_amdgcn_global_load_async_to_lds_b128(
      (__attribute__((address_space(1))) v4i*)(unsigned long long)g,
      (__attribute__((address_space(3))) v4i*)l, 0, 0);
}
__device__ __forceinline__ void async_wait0() {
#if __has_builtin(__builtin_amdgcn_s_wait_asynccnt)
  __builtin_amdgcn_s_wait_asynccnt(0);
#else
  asm volatile("s_wait_asynccnt 0x0" ::: "memory");
#endif
}
#endif

// Workspace layout (float offsets).
static const long OFF_B1P  = 0;                       // 264*512 packed W1
static const long OFF_WH1  = OFF_B1P + 264L * 512;    // N*512
static const long OFF_H1   = OFF_WH1 + (long)NN * F1; // N*512
static const long OFF_WH2  = OFF_H1  + (long)NN * F1; // N*128
static const long OFF_ES1  = OFF_WH2 + (long)NN * HID;   // H*N
static const long OFF_ED1  = OFF_ES1 + (long)NHEAD * NN;
static const long OFF_MAX1 = OFF_ED1 + (long)NHEAD * NN;
static const long OFF_SUM1 = OFF_MAX1 + (long)NHEAD * NN;
static const long OFF_ATT1 = OFF_SUM1 + (long)NHEAD * NN; // H*E
static const long OFF_ES2  = OFF_ATT1 + (long)NHEAD * EE;
static const long OFF_ED2  = OFF_ES2 + NN;
static const long OFF_MAX2 = OFF_ED2 + NN;
static const long OFF_SUM2 = OFF_MAX2 + NN;
static const long OFF_BN1S = OFF_SUM2 + NN;       // contiguous small-acc block:
static const long OFF_BN1Q = OFF_BN1S + F1;       // bn1 sum/sumsq, bn2 sum/sumsq,
static const long OFF_BN2S = OFF_BN1Q + F1;       // pool sums, counts
static const long OFF_BN2Q = OFF_BN2S + HID;
static const long OFF_POOL = OFF_BN2Q + HID;
static const long OFF_CNT  = OFF_POOL + (long)NGRAPH * HID;
static const long SMALL_ACC_LEN = F1 + F1 + HID + HID + (long)NGRAPH * HID + NGRAPH;

// ---------------------------------------------------------------------------
__global__ void zero_f32(float* p, long n) {
  long i = (long)blockIdx.x * blockDim.x + threadIdx.x;
  long stride = (long)gridDim.x * blockDim.x;
  for (; i < n; i += stride) p[i] = 0.f;
}

// Pack W1 (H, FIN, HID) -> B1p (264 x 512): B1p[k][h*128+o] = W1[h][k][o]
__global__ void pack_w1(const float* __restrict__ W1, float* __restrict__ B1p) {
  int i = blockIdx.x * 256 + threadIdx.x;
  if (i >= 264 * 512) return;
  int k = i >> 9, j = i & 511;
  int h = j >> 7, o = j & 127;
  B1p[i] = (k < FIN) ? W1[((long)h * FIN + k) * HID + o] : 0.f;
}

// ---------------------------------------------------------------------------
// Tiled fp32 WMMA GEMM: C(MxNc) = A(MxK) @ B(KxNc).
// Block = 256 threads = 8 waves, macro tile 64x32, wave grid 4x2, K-slab 32.
// Double-buffered LDS; async DMA staging overlaps slab s+1 with compute of s.
// Out-of-range rows are clamped (pollute only unstored C rows); the K
// remainder is handled by a final zero-padded synchronous slab.
// Fragment layout per CDNA5 ISA 7.12.2 (f32 16x4 A, 4x16 B, 16x16 C).
// ---------------------------------------------------------------------------
__global__ __launch_bounds__(256) void wmma_gemm_f32(
    const float* __restrict__ A, const float* __restrict__ B, float* __restrict__ C,
    int M, int K, int lda, int ldb, int ldc)
{
  __shared__ float sA[2][64][33];
  __shared__ float sB[2][32][33];
  const int tid  = threadIdx.x;
  const int wave = tid >> 5, lane = tid & 31;
  const int wr = wave >> 1, wc = wave & 1;       // 4x2 wave grid
  const int half = lane >> 4, l16 = lane & 15;
  const int row0 = blockIdx.y * 64;
  const int col0 = blockIdx.x * 32;
  v8f acc = {};
  const int K32 = K & ~31;
  const int nslab = K32 >> 5;

#if ASYNC_LDS
  auto stage_async = [&](int s, int buf) {
    const int k0 = s << 5;
    // A slab: 64x32 floats = 512 x b128 chunks, 2 per thread
    #pragma unroll
    for (int j = 0; j < 2; ++j) {
      int t = tid + j * 256;
      int r = t >> 2, seg = (t & 3) << 2;
      int gr = row0 + r; if (gr >= M) gr = M - 1;   // clamp: row-garbage is unstored
      async_cp16(A + (long)gr * lda + k0 + seg, &sA[buf][r][seg]);
    }
    // B slab: 32x32 floats = 256 chunks, 1 per thread
    {
      int r = tid >> 3, seg = (tid & 7) << 2;
      async_cp16(B + (long)(k0 + r) * ldb + col0 + seg, &sB[buf][r][seg]);
    }
  };
  if (nslab > 0) stage_async(0, 0);
  for (int s = 0; s < nslab; ++s) {
    async_wait0();
    __syncthreads();
    if (s + 1 < nslab) stage_async(s + 1, (s + 1) & 1);
    const int buf = s & 1;
    #pragma unroll
    for (int kk = 0; kk < 32; kk += 4) {
      const int ka = kk + half * 2;                // lanes 16-31 take K+2,K+3
      v2f a, b;
      a.x = sA[buf][wr * 16 + l16][ka];
      a.y = sA[buf][wr * 16 + l16][ka + 1];
      b.x = sB[buf][ka][wc * 16 + l16];
      b.y = sB[buf][ka + 1][wc * 16 + l16];
      acc = __builtin_amdgcn_wmma_f32_16x16x4_f32(
          false, a, false, b, (short)0, acc, false, false);
    }
  }
#else
  for (int s = 0; s < nslab; ++s) {
    const int k0 = s << 5;
    for (int i = tid; i < 64 * 32; i += 256) {
      int r = i >> 5, c = i & 31;
      int gr = row0 + r;
      sA[0][r][c] = (gr < M) ? A[(long)gr * lda + k0 + c] : 0.f;
    }
    for (int i = tid; i < 32 * 32; i += 256) {
      int r = i >> 5, c = i & 31;
      sB[0][r][c] = B[(long)(k0 + r) * ldb + col0 + c];
    }
    __syncthreads();
    #pragma unroll
    for (int kk = 0; kk < 32; kk += 4) {
      const int ka = kk + half * 2;
      v2f a, b;
      a.x = sA[0][wr * 16 + l16][ka];
      a.y = sA[0][wr * 16 + l16][ka + 1];
      b.x = sB[0][ka][wc * 16 + l16];
      b.y = sB[0][ka + 1][wc * 16 + l16];
      acc = __builtin_amdgcn_wmma_f32_16x16x4_f32(
          false, a, false, b, (short)0, acc, false, false);
    }
    __syncthreads();
  }
#endif

  // K remainder: zero-padded synchronous slab (K-garbage would corrupt C).
  const int rem = K - K32;
  if (rem > 0) {
    const int buf = (nslab & 1);
    for (int i = tid; i < 64 * 32; i += 256) {
      int r = i >> 5, c = i & 31;
      int gr = row0 + r, gk = K32 + c;
      sA[buf][r][c] = (gr < M && gk < K) ? A[(long)gr * lda + gk] : 0.f;
    }
    for (int i = tid; i < 32 * 32; i += 256) {
      int r = i >> 5, c = i & 31;
      int gk = K32 + r;
      sB[buf][r][c] = (gk < K) ? B[(long)gk * ldb + col0 + c] : 0.f;
    }
    __syncthreads();
    const int kpad = (rem + 3) & ~3;
    for (int kk = 0; kk < kpad; kk += 4) {
      const int ka = kk + half * 2;
      v2f a, b;
      a.x = sA[buf][wr * 16 + l16][ka];
      a.y = sA[buf][wr * 16 + l16][ka + 1];
      b.x = sB[buf][ka][wc * 16 + l16];
      b.y = sB[buf][ka + 1][wc * 16 + l16];
      acc = __builtin_amdgcn_wmma_f32_16x16x4_f32(
          false, a, false, b, (short)0, acc, false, false);
    }
  }

  const int col = col0 + wc * 16 + l16;
  #pragma unroll
  for (int i = 0; i < 8; ++i) {                  // C VGPR i -> rows i, i+8
    int row = row0 + wr * 16 + i + half * 8;
    if (row < M) C[(long)row * ldc + col] = acc[i];
  }
}

// ---------------------------------------------------------------------------
// Per-(node,head) attention scores: es = Wh[n,h,:].a_src[h], ed likewise.
// One wave32 per (node,head), shfl_xor reduction.
// ---------------------------------------------------------------------------
__global__ __launch_bounds__(256) void gat_scores(
    const float* __restrict__ Wh, const float* __restrict__ a_src,
    const float* __restrict__ a_dst, float* __restrict__ es, float* __restrict__ ed,
    int N, int Hh, int F)
{
  int wave = (int)((blockIdx.x * (long)blockDim.x + threadIdx.x) >> 5);
  int lane = threadIdx.x & 31;
  int n = wave / Hh, h = wave % Hh;
  if (n >= N) return;
  const float* row = Wh + (long)n * (Hh * F) + h * F;
  const float* as = a_src + h * F;
  const float* ad = a_dst + h * F;
  float ss = 0.f, sd = 0.f;
  for (int j = lane; j < F; j += 32) {
    float v = row[j];
    ss += v * as[j];
    sd += v * ad[j];
  }
  for (int off = 16; off > 0; off >>= 1) {
    ss += __shfl_xor(ss, off);
    sd += __shfl_xor(sd, off);
  }
  if (lane == 0) { es[(long)h * N + n] = ss; ed[(long)h * N + n] = sd; }
}

__device__ __forceinline__ float leaky02(float v) { return v > 0.f ? v : 0.2f * v; }

// Pass A: seg-max per dst (reference clamps max at 0 -> only positive values
// matter, so int-reinterpret atomicMax on a zero-initialized buffer is exact).
__global__ void edge_max(const float* __restrict__ es, const float* __restrict__ ed,
                         const int* __restrict__ src, const int* __restrict__ dst,
                         float* __restrict__ maxbuf, int E, int N, int Hh)
{
  long idx = (long)blockIdx.x * 256 + threadIdx.x;
  if (idx >= (long)E * Hh) return;
  int h = (int)(idx / E), e = (int)(idx % E);
  int d = dst[e];
  float lre = leaky02(es[(long)h * N + src[e]] + ed[(long)h * N + d]);
  if (lre > 0.f)
    atomicMax((int*)(maxbuf + (long)h * N + d), __float_as_int(lre));
}

// Pass B: exp(e - max) and seg-sum.
__global__ void edge_exp(const float* __restrict__ es, const float* __restrict__ ed,
                         const int* __restrict__ src, const int* __restrict__ dst,
                         const float* __restrict__ maxbuf, float* __restrict__ sumbuf,
                         float* __restrict__ raw, int E, int N, int Hh)
{
  long idx = (long)blockIdx.x * 256 + threadIdx.x;
  if (idx >= (long)E * Hh) return;
  int h = (int)(idx / E), e = (int)(idx % E);
  int d = dst[e];
  float lre = leaky02(es[(long)h * N + src[e]] + ed[(long)h * N + d]);
  float ex = expf(lre - maxbuf[(long)h * N + d]);
  raw[(long)h * E + e] = ex;
  atomicAdd(sumbuf + (long)h * N + d, ex);
}

// Pass C: normalize in place.
__global__ void edge_norm(const int* __restrict__ dst, const float* __restrict__ sumbuf,
                          float* __restrict__ raw, int E, int N, int Hh)
{
  long idx = (long)blockIdx.x * 256 + threadIdx.x;
  if (idx >= (long)E * Hh) return;
  int h = (int)(idx / E), e = (int)(idx % E);
  raw[idx] = raw[idx] / (sumbuf[(long)h * N + dst[e]] + 1e-8f);
}

// Message scatter: out[dst,f] += attn[h,e] * Wh[src,f] (f32 atomics, dest in L2).
__global__ void scatter_msgs(const float* __restrict__ attn, const float* __restrict__ Wh,
                             const int* __restrict__ src, const int* __restrict__ dst,
                             float* __restrict__ out, int E, int Ftot, int headF)
{
  long idx = (long)blockIdx.x * 256 + threadIdx.x;
  if (idx >= (long)E * Ftot) return;
  int e = (int)(idx / Ftot), f = (int)(idx % Ftot);
  int h = f / headF;
  float a = attn[(long)h * E + e];
  atomicAdd(out + (long)dst[e] * Ftot + f, a * Wh[(long)src[e] * Ftot + f]);
}

// ELU in place + per-column sum / sumsq (blockDim.x == F).
__global__ __launch_bounds__(512) void elu_stats(float* __restrict__ buf,
                                                 float* __restrict__ s1, float* __restrict__ s2,
                                                 int N, int F, int rowsPerBlock)
{
  int col = threadIdx.x;
  int r0 = blockIdx.x * rowsPerBlock;
  int r1 = r0 + rowsPerBlock; if (r1 > N) r1 = N;
  float a = 0.f, b = 0.f;
  for (int r = r0; r < r1; ++r) {
    long i = (long)r * F + col;
    float x = buf[i];
    float el = x > 0.f ? x : expm1f(x);
    buf[i] = el;
    a += el; b += el * el;
  }
  atomicAdd(s1 + col, a);
  atomicAdd(s2 + col, b);
}

__global__ void bn_apply(float* __restrict__ buf, const float* __restrict__ s1,
                         const float* __restrict__ s2, const float* __restrict__ g,
                         const float* __restrict__ bta, int N, int F)
{
  long idx = (long)blockIdx.x * 256 + threadIdx.x;
  if (idx >= (long)N * F) return;
  int f = (int)(idx % F);
  float mu = s1[f] / N;
  float var = s2[f] / N - mu * mu;
  buf[idx] = (buf[idx] - mu) * rsqrtf(var + 1e-5f) * g[f] + bta[f];
}

// ---------------------------------------------------------------------------
// Pooling + classifier head.
// ---------------------------------------------------------------------------
__global__ void pool_count(const int* __restrict__ batch, float* __restrict__ cnt, int N) {
  int i = blockIdx.x * 256 + threadIdx.x;
  if (i < N) atomicAdd(cnt + batch[i], 1.f);
}

__global__ void pool_sum(const float* __restrict__ h2, const int* __restrict__ batch,
                         float* __restrict__ pool, int N) {
  long idx = (long)blockIdx.x * 256 + threadIdx.x;
  if (idx >= (long)N * HID) return;
  int n = (int)(idx >> 7), f = (int)(idx & 127);
  atomicAdd(pool + (long)batch[n] * HID + f, h2[idx]);
}

__global__ __launch_bounds__(1024) void classifier(
    const float* __restrict__ pool, const float* __restrict__ cnt,
    const float* __restrict__ w1, const float* __restrict__ b1,
    const float* __restrict__ w2, const float* __restrict__ b2,
    float* __restrict__ logits)
{
  __shared__ float rep[NGRAPH][HID];
  __shared__ float hid[NGRAPH][64];
  int tid = threadIdx.x;
  for (int i = tid; i < NGRAPH * HID; i += 1024) {
    int g = i >> 7;
    float c = cnt[g]; if (c < 1.f) c = 1.f;
    rep[g][i & 127] = pool[i] / c;
  }
  __syncthreads();
  int g = tid >> 6, j = tid & 63;
  float s = b1[j];
  for (int k = 0; k < HID; ++k) s += rep[g][k] * w1[k * 64 + j];
  hid[g][j] = s > 0.f ? s : 0.f;
  __syncthreads();
  if (tid < NGRAPH * 2) {
    int gg = tid >> 1, c = tid & 1;
    float sv = b2[c];
    for (int i = 0; i < 64; ++i) sv += hid[gg][i] * w2[i * 2 + c];
    logits[gg * 2 + c] = sv;
  }
}

// ---------------------------------------------------------------------------
// Edge-importance MLP with WMMA:
//   feat(e) = [h2[src] | h2[dst]] (K=256); hidden = relu(feat @ ep_w1 + b1);
//   importance = sigmoid(hidden . ep_w2 + b2).
// Block = 256 threads = 8 waves, 16 edges/block; wave w owns hidden cols
// [16w,16w+16). ep_w1 staged per 64-K slab in LDS (async double-buffered when
// available); hidden never touches HBM.
// ---------------------------------------------------------------------------
__global__ __launch_bounds__(256) void edge_mlp(
    const float* __restrict__ h2, const int* __restrict__ src, const int* __restrict__ dst,
    const float* __restrict__ w1, const float* __restrict__ b1,
    const float* __restrict__ w2, const float* __restrict__ b2v,
    float* __restrict__ imp, int E)
{
  __shared__ float sA[16][257];      // 16 edges x 256 features
  __shared__ float sB[2][64][129];   // K-slab x 128 hidden cols (dbl-buffered)
  __shared__ float red[16];
  const int tid = threadIdx.x, wave = tid >> 5, lane = tid & 31;
  const int half = lane >> 4, l16 = lane & 15;
  const int e0 = blockIdx.x * 16;
  const int colB = wave * 16;
  if (tid < 16) red[tid] = 0.f;
  v8f acc = {};

#if ASYNC_LDS
  // async gather of edge features: 1024 chunks, 4 per thread
  #pragma unroll
  for (int j = 0; j < 4; ++j) {
    int t = tid + j * 256;
    int r = t >> 6, kc = (t & 63) << 2;
    int e = e0 + r; if (e >= E) e = E - 1;        // clamp: unstored rows
    int node = (kc < HID) ? src[e] : dst[e];
    async_cp16(h2 + (long)node * HID + (kc & 127), &sA[r][kc]);
  }
  // stage w1 slab 0
  #pragma unroll
  for (int j = 0; j < 8; ++j) {
    int t = tid + j * 256;
    int r = t >> 5, seg = (t & 31) << 2;
    async_cp16(w1 + (long)r * HID + seg, &sB[0][r][seg]);
  }
  for (int s = 0; s < 4; ++s) {
    async_wait0();
    __syncthreads();
    if (s + 1 < 4) {
      const int bufn = (s + 1) & 1, k0n = (s + 1) << 6;
      #pragma unroll
      for (int j = 0; j < 8; ++j) {
        int t = tid + j * 256;
        int r = t >> 5, seg = (t & 31) << 2;
        async_cp16(w1 + (long)(k0n + r) * HID + seg, &sB[bufn][r][seg]);
      }
    }
    const int buf = s & 1, k0 = s << 6;
    #pragma unroll
    for (int kk = 0; kk < 64; kk += 4) {
      const int ka = kk + half * 2;
      v2f a, b;
      a.x = sA[l16][k0 + ka];
      a.y = sA[l16][k0 + ka + 1];
      b.x = sB[buf][ka][colB + l16];
      b.y = sB[buf][ka + 1][colB + l16];
      acc = __builtin_amdgcn_wmma_f32_16x16x4_f32(
          false, a, false, b, (short)0, acc, false, false);
    }
  }
#else
  for (int i = tid; i < 16 * 256; i += 256) {
    int r = i >> 8, k = i & 255;
    int e = e0 + r;
    float v = 0.f;
    if (e < E) {
      int node = (k < HID) ? src[e] : dst[e];
      v = h2[(long)node * HID + (k & 127)];
    }
    sA[r][k] = v;
  }
  for (int s = 0; s < 4; ++s) {
    const int k0 = s << 6;
    __syncthreads();  // covers sA/red init on first pass, sB reuse afterwards
    for (int i = tid; i < 64 * 128; i += 256) {
      int r = i >> 7, c = i & 127;
      sB[0][r][c] = w1[(long)(k0 + r) * HID + c];
    }
    __syncthreads();
    #pragma unroll
    for (int kk = 0; kk < 64; kk += 4) {
      const int ka = kk + half * 2;
      v2f a, b;
      a.x = sA[l16][k0 + ka];
      a.y = sA[l16][k0 + ka + 1];
      b.x = sB[0][ka][colB + l16];
      b.y = sB[0][ka + 1][colB + l16];
      acc = __builtin_amdgcn_wmma_f32_16x16x4_f32(
          false, a, false, b, (short)0, acc, false, false);
    }
  }
#endif

  const float bias = b1[colB + l16];
  const float wv = w2[colB + l16];
  #pragma unroll
  for (int i = 0; i < 8; ++i) {                 // C VGPR i -> edge rows i, i+8
    float hv = acc[i] + bias;
    hv = hv > 0.f ? hv : 0.f;
    atomicAdd(&red[i + half * 8], hv * wv);
  }
  __syncthreads();
  if (tid < 16) {
    int e = e0 + tid;
    if (e < E) {
      float s = red[tid] + b2v[0];
      imp[e] = 1.f / (1.f + expf(-s));
    }
  }
}

// ---------------------------------------------------------------------------
static inline int blks(long n, int t) { return (int)((n + t - 1) / t); }

extern "C" void kernel_launch(void* const* d_in, const int* in_sizes, int n_in,
                              void* d_out, int out_size, void* d_ws, size_t ws_size,
                              hipStream_t stream)
{
  const float* x      = (const float*)d_in[0];
  const int*   ei     = (const int*)d_in[1];
  const int*   batch  = (const int*)d_in[2];
  const float* W1     = (const float*)d_in[3];
  const float* a_src1 = (const float*)d_in[4];
  const float* a_dst1 = (const float*)d_in[5];
  const float* W2     = (const float*)d_in[6];
  const float* a_src2 = (const float*)d_in[7];
  const float* a_dst2 = (const float*)d_in[8];
  const float* bn1_g  = (const float*)d_in[9];
  const float* bn1_b  = (const float*)d_in[10];
  const float* bn2_g  = (const float*)d_in[11];
  const float* bn2_b  = (const float*)d_in[12];
  const float* cls_w1 = (const float*)d_in[13];
  const float* cls_b1 = (const float*)d_in[14];
  const float* cls_w2 = (const float*)d_in[15];
  const float* cls_b2 = (const float*)d_in[16];
  const float* ep_w1  = (const float*)d_in[17];
  const float* ep_b1  = (const float*)d_in[18];
  const float* ep_w2  = (const float*)d_in[19];
  const float* ep_b2  = (const float*)d_in[20];

  const int N = in_sizes[0] / FIN;   // 50000
  const int E = in_sizes[1] / 2;     // 400000
  const int* src = ei;
  const int* dst = ei + E;

  float* ws = (float*)d_ws;
  float* out_logits = (float*)d_out;
  float* out_h2     = out_logits + NGRAPH * 2;          // N*128
  float* out_imp    = out_h2 + (long)N * HID;           // E
  float* out_attn   = out_imp + E;                      // E

  // ---- zero accumulators -------------------------------------------------
  zero_f32<<<4096, 256, 0, stream>>>(ws + OFF_H1, (long)N * F1);
  zero_f32<<<1024, 256, 0, stream>>>(ws + OFF_MAX1, 2L * NHEAD * N);  // max1+sum1
  zero_f32<<<512, 256, 0, stream>>>(ws + OFF_MAX2, 2L * N);           // max2+sum2
  zero_f32<<<16, 256, 0, stream>>>(ws + OFF_BN1S, SMALL_ACC_LEN);
  zero_f32<<<4096, 256, 0, stream>>>(out_h2, (long)N * HID);

  // ---- layer 1: Wh1 = x @ W1 (heads fused into 512 cols) -----------------
  pack_w1<<<blks(264L * 512, 256), 256, 0, stream>>>(W1, ws + OFF_B1P);
  {
    dim3 g(F1 / 32, blks(N, 64));
    wmma_gemm_f32<<<g, 256, 0, stream>>>(x, ws + OFF_B1P, ws + OFF_WH1,
                                         N, FIN, FIN, F1, F1);
  }
  gat_scores<<<blks((long)N * NHEAD * 32, 256), 256, 0, stream>>>(
      ws + OFF_WH1, a_src1, a_dst1, ws + OFF_ES1, ws + OFF_ED1, N, NHEAD, HID);

  edge_max<<<blks((long)E * NHEAD, 256), 256, 0, stream>>>(
      ws + OFF_ES1, ws + OFF_ED1, src, dst, ws + OFF_MAX1, E, N, NHEAD);
  edge_exp<<<blks((long)E * NHEAD, 256), 256, 0, stream>>>(
      ws + OFF_ES1, ws + OFF_ED1, src, dst, ws + OFF_MAX1, ws + OFF_SUM1,
      ws + OFF_ATT1, E, N, NHEAD);
  edge_norm<<<blks((long)E * NHEAD, 256), 256, 0, stream>>>(
      dst, ws + OFF_SUM1, ws + OFF_ATT1, E, N, NHEAD);
  scatter_msgs<<<blks((long)E * F1, 256), 256, 0, stream>>>(
      ws + OFF_ATT1, ws + OFF_WH1, src, dst, ws + OFF_H1, E, F1, HID);

  elu_stats<<<blks(N, 128), F1, 0, stream>>>(ws + OFF_H1, ws + OFF_BN1S,
                                             ws + OFF_BN1Q, N, F1, 128);
  bn_apply<<<blks((long)N * F1, 256), 256, 0, stream>>>(
      ws + OFF_H1, ws + OFF_BN1S, ws + OFF_BN1Q, bn1_g, bn1_b, N, F1);

  // ---- layer 2: Wh2 = h1 @ W2 --------------------------------------------
  {
    dim3 g(HID / 32, blks(N, 64));
    wmma_gemm_f32<<<g, 256, 0, stream>>>(ws + OFF_H1, W2, ws + OFF_WH2,
                                         N, F1, F1, HID, HID);
  }
  gat_scores<<<blks((long)N * 32, 256), 256, 0, stream>>>(
      ws + OFF_WH2, a_src2, a_dst2, ws + OFF_ES2, ws + OFF_ED2, N, 1, HID);

  edge_max<<<blks((long)E, 256), 256, 0, stream>>>(
      ws + OFF_ES2, ws + OFF_ED2, src, dst, ws + OFF_MAX2, E, N, 1);
  edge_exp<<<blks((long)E, 256), 256, 0, stream>>>(
      ws + OFF_ES2, ws + OFF_ED2, src, dst, ws + OFF_MAX2, ws + OFF_SUM2,
      out_attn, E, N, 1);
  edge_norm<<<blks((long)E, 256), 256, 0, stream>>>(
      dst, ws + OFF_SUM2, out_attn, E, N, 1);
  scatter_msgs<<<blks((long)E * HID, 256), 256, 0, stream>>>(
      out_attn, ws + OFF_WH2, src, dst, out_h2, E, HID, HID);

  elu_stats<<<blks(N, 512), HID, 0, stream>>>(out_h2, ws + OFF_BN2S,
                                              ws + OFF_BN2Q, N, HID, 512);
  bn_apply<<<blks((long)N * HID, 256), 256, 0, stream>>>(
      out_h2, ws + OFF_BN2S, ws + OFF_BN2Q, bn2_g, bn2_b, N, HID);

  // ---- pooling + classifier ----------------------------------------------
  pool_count<<<blks(N, 256), 256, 0, stream>>>(batch, ws + OFF_CNT, N);
  pool_sum<<<blks((long)N * HID, 256), 256, 0, stream>>>(out_h2, batch,
                                                         ws + OFF_POOL, N);
  classifier<<<1, 1024, 0, stream>>>(ws + OFF_POOL, ws + OFF_CNT,
                                     cls_w1, cls_b1, cls_w2, cls_b2, out_logits);

  // ---- edge importance MLP (WMMA) ----------------------------------------
  edge_mlp<<<blks(E, 16), 256, 0, stream>>>(out_h2, src, dst, ep_w1, ep_b1,
                                            ep_w2, ep_b2, out_imp, E);

  (void)n_in; (void)out_size; (void)ws_size;
}